// ComplexCoordAttention_6631429505705
// MI455X (gfx1250) — compile-verified
//
#include <hip/hip_runtime.h>
#include <hip/hip_bf16.h>
#include <math.h>

#define B_   8
#define S_   64
#define P_   64
#define NL_  128
#define NP_  256
#define NT_  384   // NL + NP
#define EPS_ 1e-6f

typedef __attribute__((ext_vector_type(2))) float v2f;
typedef __attribute__((ext_vector_type(4))) float f4;
typedef __attribute__((ext_vector_type(8))) float v8f;

// ------------------------------------------------------------------
// Kernel 1: per-(b,s) normalization scale = set_weights[s]/(mean_len+EPS)
// ------------------------------------------------------------------
__global__ void k_norm_scale(const float* __restrict__ lig,
                             const float* __restrict__ pro,
                             const float* __restrict__ ligm,
                             const float* __restrict__ prom,
                             const float* __restrict__ setw,
                             float* __restrict__ scale) {
    int t = blockIdx.x * blockDim.x + threadIdx.x;
    if (t >= B_ * S_) return;
    int s = t % S_;
    float total = 0.f, natoms = 0.f;
    {
        const float* c = lig + (size_t)t * NL_ * 3;
        const float* m = ligm + (size_t)t * NL_;
        for (int n = 0; n < NL_; ++n) {
            float x = c[n * 3 + 0], y = c[n * 3 + 1], z = c[n * 3 + 2];
            float sq = x * x + y * y + z * z;
            float len = sq > 0.f ? sqrtf(sq) : 0.f;
            total += len * m[n];
            natoms += m[n];
        }
    }
    {
        const float* c = pro + (size_t)t * NP_ * 3;
        const float* m = prom + (size_t)t * NP_;
        for (int n = 0; n < NP_; ++n) {
            float x = c[n * 3 + 0], y = c[n * 3 + 1], z = c[n * 3 + 2];
            float sq = x * x + y * y + z * z;
            float len = sq > 0.f ? sqrtf(sq) : 0.f;
            total += len * m[n];
            natoms += m[n];
        }
    }
    float mean = total / natoms;
    scale[t] = setw[s] / (mean + EPS_);
}

// ------------------------------------------------------------------
// A-element fetch for projection GEMM: scaled, masked coordinate.
// A[M=n, K=s] = coord[b,s,n,c] * mask[b,s,n] * scale[b,s]
// ------------------------------------------------------------------
__device__ __forceinline__ float fetchA(const float* __restrict__ lig,
                                        const float* __restrict__ pro,
                                        const float* __restrict__ ligm,
                                        const float* __restrict__ prom,
                                        const float* __restrict__ scale,
                                        int b, int s, int M, int c) {
    float sc = scale[b * S_ + s];
    if (M < NL_) {
        size_t base = (size_t)(b * S_ + s) * NL_ + M;
        return lig[base * 3 + c] * ligm[base] * sc;
    } else {
        size_t base = (size_t)(b * S_ + s) * NP_ + (M - NL_);
        return pro[base * 3 + c] * prom[base] * sc;
    }
}

// ------------------------------------------------------------------
// Kernel 2: projection GEMM via V_WMMA_F32_16X16X4_F32.
// proj[b,c,n,p] = sum_s coord_scaled[b,s,n,c] * w_sel[p,s]
// One wave (32 lanes) per 16x16 tile; K=S=64 -> 16 wmma steps.
// ------------------------------------------------------------------
__global__ void __launch_bounds__(32) k_proj_wmma(
        const float* __restrict__ lig, const float* __restrict__ pro,
        const float* __restrict__ ligm, const float* __restrict__ prom,
        const float* __restrict__ ligw, const float* __restrict__ prow,
        const float* __restrict__ scale, float* __restrict__ proj) {
    int t = blockIdx.x;
    int ptile = t & 3;  t >>= 2;                  // P_/16 = 4
    int ntile = t % (NT_ / 16); t /= (NT_ / 16);  // 24
    int c = t % 3;
    int b = t / 3;

    int lane  = threadIdx.x;
    int mrow  = lane & 15;
    int khalf = (lane >> 4) << 1;         // 0 for lanes 0-15, 2 for lanes 16-31
    int M = ntile * 16 + mrow;            // A-matrix row (n)
    int p = ptile * 16 + mrow;            // B-matrix column (p)
    const float* w = (ntile < (NL_ / 16)) ? ligw : prow;  // tile is homogeneous in n

    v8f acc = {};
    for (int k0 = 0; k0 < S_; k0 += 4) {
        int sA = k0 + khalf;
        v2f a, bf;
        a.x  = fetchA(lig, pro, ligm, prom, scale, b, sA,     M, c);
        a.y  = fetchA(lig, pro, ligm, prom, scale, b, sA + 1, M, c);
        bf.x = w[p * S_ + sA];
        bf.y = w[p * S_ + sA + 1];
        acc = __builtin_amdgcn_wmma_f32_16x16x4_f32(
            false, a, false, bf, (short)0, acc, false, false);
    }

    int m0 = ntile * 16, p0 = ptile * 16;
    int hi = (lane >= 16) ? 8 : 0;
    int pc = p0 + (lane & 15);
    for (int r = 0; r < 8; ++r) {
        int Mr = m0 + r + hi;
        proj[((size_t)(b * 3 + c) * NT_ + Mr) * P_ + pc] = acc[r];
    }
}

// ------------------------------------------------------------------
// Kernel 3: online-softmax attention + direction-weighted reduction.
// One workgroup per (b, i): 256 threads = 16 n-slices x 16 p-groups.
// Each thread owns 4 consecutive p's -> all streaming loads are b128.
// Single pass over messages (the HBM-roofline tensor, read once),
// with global_prefetch issued 6 rows ahead.
// Produces upd[b,c,i,p] = (sum_n attn*norm_dist_c) * sqrt(sum_n attn^2)
// ------------------------------------------------------------------
__global__ void __launch_bounds__(256) k_attn(
        const float* __restrict__ messages, const int* __restrict__ adj,
        const float* __restrict__ proj, float* __restrict__ upd) {
    int b = blockIdx.x / NL_;
    int i = blockIdx.x % NL_;
    int tid = threadIdx.x;
    int pg = tid & 15;        // p-group: p = pg*4 .. pg*4+3
    int ny = tid >> 4;        // 16 n-slices of NT/16 = 24 rows each

    __shared__ int   s_adj[NT_];
    __shared__ float s_m[16][64], s_d[16][64], s_s2[16][64];
    __shared__ float s_a0[16][64], s_a1[16][64], s_a2[16][64];

    const int* adjRow = adj + (size_t)(b * NL_ + i) * NT_;
    for (int n = tid; n < NT_; n += 256) s_adj[n] = adjRow[n];
    __syncthreads();

    const f4* projB = (const f4*)(proj + (size_t)b * 3 * NT_ * P_);
    f4 pi0 = projB[(0 * NT_ + i) * (P_ / 4) + pg];
    f4 pi1 = projB[(1 * NT_ + i) * (P_ / 4) + pg];
    f4 pi2 = projB[(2 * NT_ + i) * (P_ / 4) + pg];

    const f4* msg4 = (const f4*)(messages + (size_t)(b * NL_ + i) * NT_ * P_);

    float m[4], d[4], s2[4], a0[4], a1[4], a2[4];
#pragma unroll
    for (int j = 0; j < 4; ++j) {
        m[j] = -INFINITY; d[j] = 0.f; s2[j] = 0.f;
        a0[j] = 0.f; a1[j] = 0.f; a2[j] = 0.f;
    }

    const int CH = NT_ / 16;   // 24 rows per slice
    int n0 = ny * CH;
    for (int n = n0; n < n0 + CH; ++n) {
        int npf = n + 6;
        if (npf < n0 + CH)
            __builtin_prefetch(&msg4[(size_t)npf * (P_ / 4) + pg], 0, 0);
        if (s_adj[n] <= 0) continue;      // attn_mask = -inf -> exp = 0
        f4 x  = msg4[(size_t)n * (P_ / 4) + pg];
        f4 q0 = projB[(0 * NT_ + n) * (P_ / 4) + pg];
        f4 q1 = projB[(1 * NT_ + n) * (P_ / 4) + pg];
        f4 q2 = projB[(2 * NT_ + n) * (P_ / 4) + pg];
#pragma unroll
        for (int j = 0; j < 4; ++j) {
            float xv = x[j];
            float w;
            if (xv > m[j]) {
                float r = __expf(m[j] - xv);
                d[j] *= r; a0[j] *= r; a1[j] *= r; a2[j] *= r; s2[j] *= r * r;
                m[j] = xv; w = 1.f;
            } else {
                w = __expf(xv - m[j]);
            }
            float v0 = pi0[j] - q0[j];
            float v1 = pi1[j] - q1[j];
            float v2 = pi2[j] - q2[j];
            float sq = v0 * v0 + v1 * v1 + v2 * v2;
            float len = sq > 0.f ? sqrtf(sq) : 0.f;
            float inv = 1.f / (len + EPS_);
            d[j]  += w;
            s2[j] += w * w;
            a0[j] += w * v0 * inv;
            a1[j] += w * v1 * inv;
            a2[j] += w * v2 * inv;
        }
    }
#pragma unroll
    for (int j = 0; j < 4; ++j) {
        int p = pg * 4 + j;
        s_m[ny][p] = m[j];  s_d[ny][p] = d[j];  s_s2[ny][p] = s2[j];
        s_a0[ny][p] = a0[j]; s_a1[ny][p] = a1[j]; s_a2[ny][p] = a2[j];
    }
    __syncthreads();

    if (tid < 64) {
        int p = tid;
        float M = -INFINITY, D = 0.f, S2 = 0.f, A0 = 0.f, A1 = 0.f, A2 = 0.f;
        for (int jj = 0; jj < 16; ++jj) {
            float dj = s_d[jj][p];
            if (dj == 0.f) continue;      // empty partial
            float mj = s_m[jj][p];
            float w;
            if (mj > M) {
                float r = __expf(M - mj);
                D *= r; A0 *= r; A1 *= r; A2 *= r; S2 *= r * r;
                M = mj; w = 1.f;
            } else {
                w = __expf(mj - M);
            }
            D  += w * dj;
            S2 += w * w * s_s2[jj][p];
            A0 += w * s_a0[jj][p];
            A1 += w * s_a1[jj][p];
            A2 += w * s_a2[jj][p];
        }
        // upd_c = (A_c / D) * sqrt(S2) / D
        float f = sqrtf(S2) / (D * D);
        size_t strideC = (size_t)NL_ * P_;
        size_t base = ((size_t)(b * 3) * NL_ + i) * P_ + p;
        upd[base + 0 * strideC] = A0 * f;
        upd[base + 1 * strideC] = A1 * f;
        upd[base + 2 * strideC] = A2 * f;
    }
}

// ------------------------------------------------------------------
// Kernel 4: output projection via WMMA.
// out[b,s,i,c] = sum_p upd[b,c,i,p] * attn_proj_w[s,p]
// One wave per 16x16 (i,s) tile; K=P=64 -> 16 wmma steps.
// ------------------------------------------------------------------
__global__ void __launch_bounds__(32) k_out_wmma(
        const float* __restrict__ upd, const float* __restrict__ attnw,
        float* __restrict__ out) {
    int t = blockIdx.x;
    int stile = t & 3;  t >>= 2;          // S_/16 = 4
    int itile = t & 7;  t >>= 3;          // NL_/16 = 8
    int c = t % 3;
    int b = t / 3;

    int lane  = threadIdx.x;
    int mrow  = lane & 15;
    int khalf = (lane >> 4) << 1;
    int i = itile * 16 + mrow;            // A row
    int s = stile * 16 + mrow;            // B column

    const float* Arow = upd + ((size_t)(b * 3 + c) * NL_ + i) * P_;

    v8f acc = {};
    for (int k0 = 0; k0 < P_; k0 += 4) {
        int kk = k0 + khalf;
        v2f a, bf;
        a.x  = Arow[kk];
        a.y  = Arow[kk + 1];
        bf.x = attnw[s * P_ + kk];
        bf.y = attnw[s * P_ + kk + 1];
        acc = __builtin_amdgcn_wmma_f32_16x16x4_f32(
            false, a, false, bf, (short)0, acc, false, false);
    }

    int m0 = itile * 16, s0 = stile * 16;
    int hi = (lane >= 16) ? 8 : 0;
    int sc = s0 + (lane & 15);
    for (int r = 0; r < 8; ++r) {
        int Mr = m0 + r + hi;
        out[(((size_t)b * S_ + sc) * NL_ + Mr) * 3 + c] = acc[r];
    }
}

// ------------------------------------------------------------------
extern "C" void kernel_launch(void* const* d_in, const int* in_sizes, int n_in,
                              void* d_out, int out_size, void* d_ws, size_t ws_size,
                              hipStream_t stream) {
    const float* lig   = (const float*)d_in[0];  // (B,S,Nl,3)
    const float* msg   = (const float*)d_in[1];  // (B,Nl,N,P)
    const int*   adj   = (const int*)  d_in[2];  // (B,Nl,N)
    const float* ligm  = (const float*)d_in[3];  // (B,S,Nl)
    const float* pro   = (const float*)d_in[4];  // (B,S,Np,3)
    const float* prom  = (const float*)d_in[5];  // (B,S,Np)
    const float* ligw  = (const float*)d_in[6];  // (P,S)
    const float* prow  = (const float*)d_in[7];  // (P,S)
    const float* attnw = (const float*)d_in[8];  // (S,P)
    const float* setw  = (const float*)d_in[9];  // (S,)
    float* out = (float*)d_out;                  // (B,S,Nl,3)

    float* scale = (float*)d_ws;                             // B*S      = 512
    float* proj  = scale + (size_t)B_ * S_;                  // B*3*NT*P = 589824
    float* upd   = proj + (size_t)B_ * 3 * NT_ * P_;         // B*3*NL*P = 196608

    hipLaunchKernelGGL(k_norm_scale, dim3(2), dim3(256), 0, stream,
                       lig, pro, ligm, prom, setw, scale);

    hipLaunchKernelGGL(k_proj_wmma,
                       dim3(B_ * 3 * (NT_ / 16) * (P_ / 16)), dim3(32), 0, stream,
                       lig, pro, ligm, prom, ligw, prow, scale, proj);

    hipLaunchKernelGGL(k_attn, dim3(B_ * NL_), dim3(256), 0, stream,
                       msg, adj, proj, upd);

    hipLaunchKernelGGL(k_out_wmma,
                       dim3(B_ * 3 * (NL_ / 16) * (S_ / 16)), dim3(32), 0, stream,
                       upd, attnw, out);
}